// GNN_node_1408749273896
// MI455X (gfx1250) — compile-verified
//
#include <hip/hip_runtime.h>

typedef __attribute__((ext_vector_type(2))) float v2f;
typedef __attribute__((ext_vector_type(8))) float v8f;

#define EMB 128

// ---------------- utility ----------------
__global__ void zero_f32(float* __restrict__ p, int n) {
  int i = blockIdx.x * blockDim.x + threadIdx.x;
  if (i < n) p[i] = 0.0f;
}

// h[node][:] = atom_emb[x[node]][:]   (x has 1 feature column -> pure gather)
// one wave32 per node, one float4 per lane (128 floats)
__global__ void encode_nodes(const int* __restrict__ x,
                             const float* __restrict__ atom_emb,
                             float* __restrict__ h, int n_nodes) {
  int wave = (blockIdx.x * blockDim.x + threadIdx.x) >> 5;
  int lane = threadIdx.x & 31;
  if (wave >= n_nodes) return;
  int a = x[wave];
  float4 v = ((const float4*)(atom_emb + (size_t)a * EMB))[lane];
  ((float4*)(h + (size_t)wave * EMB))[lane] = v;
}

// deg[row[e]] += 1
__global__ void degree_kernel(const int* __restrict__ row,
                              float* __restrict__ deg, int n_edges) {
  int e = blockIdx.x * blockDim.x + threadIdx.x;
  if (e < n_edges) atomicAdd(&deg[row[e]], 1.0f);
}

// in place: deg -> rsqrt(deg==0 ? 1 : deg)
__global__ void dinv_kernel(float* __restrict__ deg, int n_nodes) {
  int i = blockIdx.x * blockDim.x + threadIdx.x;
  if (i < n_nodes) {
    float d = deg[i];
    deg[i] = rsqrtf(d == 0.0f ? 1.0f : d);
  }
}

// norm[e] = dinv[row[e]] * dinv[col[e]]
__global__ void norm_kernel(const int* __restrict__ row, const int* __restrict__ col,
                            const float* __restrict__ dinv,
                            float* __restrict__ nrm, int n_edges) {
  int e = blockIdx.x * blockDim.x + threadIdx.x;
  if (e < n_edges) nrm[e] = dinv[row[e]] * dinv[col[e]];
}

// agg[col[e]][:] += norm[e] * h[row[e]][:] * bond_emb[attr[e]][:]
// one wave32 per edge, float4 per lane; bond table (8 KB) staged in LDS
__global__ void message_scatter(const float* __restrict__ h,
                                const int* __restrict__ row,
                                const int* __restrict__ col,
                                const int* __restrict__ attr,
                                const float* __restrict__ nrm,
                                const float* __restrict__ bond_emb,
                                float* __restrict__ agg, int n_edges) {
  __shared__ float be[16 * EMB];
  for (int i = threadIdx.x; i < 16 * EMB; i += blockDim.x) be[i] = bond_emb[i];
  __syncthreads();

  int wave = threadIdx.x >> 5;
  int lane = threadIdx.x & 31;
  int e = blockIdx.x * (blockDim.x >> 5) + wave;
  if (e >= n_edges) return;

  int r = row[e], c = col[e], a = attr[e];
  float w = nrm[e];
  float4 hv = ((const float4*)(h + (size_t)r * EMB))[lane];
  float4 ev = ((const float4*)(be + a * EMB))[lane];
  float* dst = agg + (size_t)c * EMB + lane * 4;
  atomicAdd(dst + 0, w * hv.x * ev.x);
  atomicAdd(dst + 1, w * hv.y * ev.y);
  atomicAdd(dst + 2, w * hv.z * ev.z);
  atomicAdd(dst + 3, w * hv.w * ev.w);
}

// out = relu(A @ W + bias), A:[n_rows x 128], W:[128 x 128], bias:[128]
// block = 256 threads = 8 waves; block owns a 16-row strip of A (LDS-staged,
// pitch 132 for bank-conflict-free fragment reads), wave w computes the 16x16
// tile at columns [16w, 16w+16) via V_WMMA_F32_16X16X4_F32, K-loop of 32.
__global__ void gemm_bias_relu_wmma(const float* __restrict__ A,
                                    const float* __restrict__ W,
                                    const float* __restrict__ bias,
                                    float* __restrict__ out, int n_rows) {
  __shared__ float lds_a[16 * 132];
  int row_base = blockIdx.x * 16;

  for (int i = threadIdx.x; i < 16 * EMB; i += 256) {
    int r = i >> 7;            // 0..15
    int k = i & 127;           // 0..127
    int gr = row_base + r;
    lds_a[r * 132 + k] = (gr < n_rows) ? A[(size_t)gr * EMB + k] : 0.0f;
  }
  __syncthreads();

  int wave = threadIdx.x >> 5;
  int lane = threadIdx.x & 31;
  int col_base = wave * 16;
  int mn   = lane & 15;        // A row / B col owned by this lane
  int half = lane >> 4;        // 0: k+{0,1}, 1: k+{2,3}
  int koff = half * 2;

  v8f acc = {};
#pragma unroll 4
  for (int kt = 0; kt < 32; ++kt) {
    int k0 = kt * 4 + koff;
    v2f a, b;
    a.x = lds_a[mn * 132 + k0];
    a.y = lds_a[mn * 132 + k0 + 1];
    b.x = W[(size_t)k0 * EMB + col_base + mn];
    b.y = W[(size_t)(k0 + 1) * EMB + col_base + mn];
    // 8 args: (neg_a, A, neg_b, B, c_mod, C, reuse_a, reuse_b)
    acc = __builtin_amdgcn_wmma_f32_16x16x4_f32(false, a, false, b,
                                                (short)0, acc, false, false);
  }

  float bv = bias[col_base + mn];
  float* base = out + (size_t)(row_base + half * 8) * EMB + col_base + mn;

  if (row_base + 16 <= n_rows) {
    // fast path: whole tile in-bounds -> 8 unconditional stores
#pragma unroll
    for (int r = 0; r < 8; ++r) {
      float v = acc[r] + bv;
      base[(size_t)r * EMB] = v > 0.0f ? v : 0.0f;
    }
  } else {
#pragma unroll
    for (int r = 0; r < 8; ++r) {
      int gr = row_base + r + half * 8;
      if (gr < n_rows) {
        float v = acc[r] + bv;
        base[(size_t)r * EMB] = v > 0.0f ? v : 0.0f;
      }
    }
  }
}

extern "C" void kernel_launch(void* const* d_in, const int* in_sizes, int n_in,
                              void* d_out, int out_size, void* d_ws, size_t ws_size,
                              hipStream_t stream) {
  const int*   x         = (const int*)d_in[0];
  const int*   edge_idx  = (const int*)d_in[1];
  const int*   edge_attr = (const int*)d_in[2];
  const float* atom_emb  = (const float*)d_in[3];
  const float* bond_emb  = (const float*)d_in[4];
  const float* lin_W     = (const float*)d_in[5];
  const float* lin_b     = (const float*)d_in[6];

  const int N = in_sizes[0];        // x is [N,1]
  const int E = in_sizes[2];        // edge_attr is [E,1]
  const int L = 4;
  const int* row = edge_idx;        // edge_index[0]
  const int* col = edge_idx + E;    // edge_index[1]

  char* ws = (char*)d_ws;
  float* h   = (float*)ws;  ws += (size_t)N * EMB * sizeof(float);
  float* agg = (float*)ws;  ws += (size_t)N * EMB * sizeof(float);
  float* deg = (float*)ws;  ws += (size_t)N * sizeof(float);     // becomes dinv
  float* nrm = (float*)ws;  ws += (size_t)E * sizeof(float);

  // encoders + GCN normalization
  encode_nodes<<<(N + 7) / 8, 256, 0, stream>>>(x, atom_emb, h, N);
  zero_f32<<<(N + 255) / 256, 256, 0, stream>>>(deg, N);
  degree_kernel<<<(E + 255) / 256, 256, 0, stream>>>(row, deg, E);
  dinv_kernel<<<(N + 255) / 256, 256, 0, stream>>>(deg, N);
  norm_kernel<<<(E + 255) / 256, 256, 0, stream>>>(row, col, deg, nrm, E);

  const int agg_elems = N * EMB;
  for (int l = 0; l < L; ++l) {
    zero_f32<<<(agg_elems + 255) / 256, 256, 0, stream>>>(agg, agg_elems);
    message_scatter<<<(E + 7) / 8, 256, 0, stream>>>(h, row, col, edge_attr,
                                                     nrm, bond_emb, agg, E);
    float* dst = (l == L - 1) ? (float*)d_out : h;
    gemm_bias_relu_wmma<<<(N + 15) / 16, 256, 0, stream>>>(
        agg, lin_W + (size_t)l * EMB * EMB, lin_b + (size_t)l * EMB, dst, N);
  }
}